// YunetPostProcessing_37606733644117
// MI455X (gfx1250) — compile-verified
//
#include <hip/hip_runtime.h>
#include <math.h>

typedef __attribute__((ext_vector_type(2))) float v2f;
typedef __attribute__((ext_vector_type(8))) float v8f;

#define VAR0f 0.1f
#define VAR1f 0.2f
#define IOU_THRf 0.3f

__device__ __forceinline__ unsigned ballot32(bool p) {
#if __has_builtin(__builtin_amdgcn_ballot_w32)
  return __builtin_amdgcn_ballot_w32(p);
#else
  return (unsigned)__ballot(p);
#endif
}

// ---------------------------------------------------------------------------
// Kernel 1: decode boxes / landmarks / scores (elementwise, trivially small)
// ---------------------------------------------------------------------------
__global__ void decode_kernel(const float* __restrict__ loc,
                              const float* __restrict__ conf,
                              const float* __restrict__ iou,
                              const float* __restrict__ priors,
                              float* __restrict__ scores,
                              float* __restrict__ bbox,
                              float* __restrict__ lm,
                              int N) {
  int i = blockIdx.x * blockDim.x + threadIdx.x;
  if (i >= N) return;
  float px = priors[i * 4 + 0], py = priors[i * 4 + 1];
  float pw = priors[i * 4 + 2], ph = priors[i * 4 + 3];
  const float* L = loc + (size_t)i * 14;
  float cx = px + L[0] * VAR0f * pw;
  float cy = py + L[1] * VAR0f * ph;
  float whx = pw * expf(L[2] * VAR0f) * 0.5f;
  float why = ph * expf(L[3] * VAR1f) * 0.5f;
  bbox[i * 4 + 0] = cx - whx;
  bbox[i * 4 + 1] = cy - why;
  bbox[i * 4 + 2] = cx + whx;
  bbox[i * 4 + 3] = cy + why;
#pragma unroll
  for (int t = 0; t < 5; ++t) {
    lm[i * 10 + 2 * t + 0] = px + L[4 + 2 * t] * VAR0f * pw;
    lm[i * 10 + 2 * t + 1] = py + L[5 + 2 * t] * VAR0f * ph;
  }
  float ic = iou[i];
  ic = fminf(fmaxf(ic, 0.0f), 1.0f);
  scores[i] = sqrtf(conf[i * 2 + 1] * ic);
}

// ---------------------------------------------------------------------------
// Kernel 2: single-workgroup bitonic sort of (~score_bits, idx) packed u64.
// Ascending u64 sort == descending score with stable (ascending-index) ties.
// LDS is dynamic (sortn*8 bytes, up to 64KB; WGP has 320KB).
// ---------------------------------------------------------------------------
__global__ __launch_bounds__(1024) void sort_kernel(const float* __restrict__ scores,
                                                    unsigned* __restrict__ order,
                                                    int N, int sortn) {
  extern __shared__ unsigned long long skey[];
  int tid = threadIdx.x;
  for (int i = tid; i < sortn; i += 1024) {
    unsigned long long k;
    if (i < N) {
      unsigned sb = ~__float_as_uint(scores[i]);  // scores >= 0 -> monotone bits
      k = (((unsigned long long)sb) << 32) | (unsigned)i;
    } else {
      k = ~0ull;  // pad to end
    }
    skey[i] = k;
  }
  __syncthreads();
  for (int k = 2; k <= sortn; k <<= 1) {
    for (int j = k >> 1; j > 0; j >>= 1) {
      for (int i = tid; i < sortn; i += 1024) {
        int ixj = i ^ j;
        if (ixj > i) {
          bool up = ((i & k) == 0);
          unsigned long long a = skey[i], b = skey[ixj];
          if ((a > b) == up) { skey[i] = b; skey[ixj] = a; }
        }
      }
      __syncthreads();
    }
  }
  for (int i = tid; i < N; i += 1024)
    order[i] = (unsigned)(skey[i] & 0xFFFFFFFFull);
}

// ---------------------------------------------------------------------------
// Kernel 3: gather sorted boxes into SoA + precompute areas
// ---------------------------------------------------------------------------
__global__ void gather_sorted_kernel(const unsigned* __restrict__ order,
                                     const float* __restrict__ bbox,
                                     float* __restrict__ x1s, float* __restrict__ y1s,
                                     float* __restrict__ x2s, float* __restrict__ y2s,
                                     float* __restrict__ areas, int N) {
  int i = blockIdx.x * blockDim.x + threadIdx.x;
  if (i >= N) return;
  unsigned o = order[i];
  float b0 = bbox[o * 4 + 0], b1 = bbox[o * 4 + 1];
  float b2 = bbox[o * 4 + 2], b3 = bbox[o * 4 + 3];
  x1s[i] = b0; y1s[i] = b1; x2s[i] = b2; y2s[i] = b3;
  areas[i] = (b2 - b0) * (b3 - b1);
}

// ---------------------------------------------------------------------------
// Kernel 4: suppression bitmask. One wave per 16-row tile; loops over all
// 32-column groups. area_m + area_n outer-sum for each 16x16 tile computed
// on the matrix unit via V_WMMA_F32_16X16X4_F32 (rank-2 matmul:
// A = [area_m, 1, 0, 0] (16x4), B = [1; area_n; 0; 0] (4x16)), consumed
// directly in the IoU denominator. Predicates packed with wave32 ballots.
// ---------------------------------------------------------------------------
__global__ __launch_bounds__(32) void iou_mask_kernel(const float* __restrict__ x1s,
                                                      const float* __restrict__ y1s,
                                                      const float* __restrict__ x2s,
                                                      const float* __restrict__ y2s,
                                                      const float* __restrict__ areas,
                                                      unsigned* __restrict__ maskw,
                                                      int N, int w32) {
  __shared__ float sx1[16], sy1[16], sx2[16], sy2[16], sar[16];
  int lane = threadIdx.x;
  int r0 = blockIdx.x * 16;
  if (lane < 16) {
    int r = min(r0 + lane, N - 1);
    sx1[lane] = x1s[r]; sy1[lane] = y1s[r];
    sx2[lane] = x2s[r]; sy2[lane] = y2s[r];
    sar[lane] = areas[r];
  }
  __syncthreads();

  int nloc = lane & 15;
  int hi = lane >> 4;  // 0: rows v, 1: rows v+8 (D-matrix lane layout)

  // A matrix (16x4, M x K): lanes 0-15 hold K=0,1 ; lanes 16-31 hold K=2,3
  v2f A;
  A.x = hi ? 0.0f : sar[nloc];  // K=0 = area_m (lanes 0-15, M=lane)
  A.y = hi ? 0.0f : 1.0f;       // K=1 = 1
  // (lanes 16-31: K=2,3 = 0)

  for (int cg = 0; cg < w32; ++cg) {
    unsigned bal[2][8];
#pragma unroll
    for (int s = 0; s < 2; ++s) {
      int cIdx = cg * 32 + s * 16 + nloc;
      int cc = min(cIdx, N - 1);
      float cx1 = x1s[cc], cy1 = y1s[cc], cx2 = x2s[cc], cy2 = y2s[cc];
      float car = areas[cc];

      // B matrix (4x16, K x N): lanes 0-15 hold K=0,1 ; lanes 16-31 hold K=2,3
      v2f B;
      B.x = hi ? 0.0f : 1.0f;  // K=0 row = ones
      B.y = hi ? 0.0f : car;   // K=1 row = area_n
      float asum[8];
#if __has_builtin(__builtin_amdgcn_wmma_f32_16x16x4_f32)
      v8f C = {0.f, 0.f, 0.f, 0.f, 0.f, 0.f, 0.f, 0.f};
      v8f D = __builtin_amdgcn_wmma_f32_16x16x4_f32(
          /*neg_a=*/false, A, /*neg_b=*/false, B,
          /*c_mod=*/(short)0, C, /*reuse_a=*/false, /*reuse_b=*/false);
#pragma unroll
      for (int v = 0; v < 8; ++v) asum[v] = D[v];
#else
#pragma unroll
      for (int v = 0; v < 8; ++v) asum[v] = sar[v + 8 * hi] + car;
#endif
#pragma unroll
      for (int v = 0; v < 8; ++v) {
        int m = v + 8 * hi;        // D layout: lanes 0-15 -> M=v, 16-31 -> M=v+8
        int rIdx = r0 + m;
        float ix1 = fmaxf(sx1[m], cx1);
        float iy1 = fmaxf(sy1[m], cy1);
        float ix2 = fminf(sx2[m], cx2);
        float iy2 = fminf(sy2[m], cy2);
        float inter = fmaxf(ix2 - ix1, 0.0f) * fmaxf(iy2 - iy1, 0.0f);
        float uni = asum[v] - inter;  // area_m + area_n - inter (WMMA result)
        bool pred = (inter > IOU_THRf * uni) && (cIdx > rIdx) &&
                    (cIdx < N) && (rIdx < N);
        bal[s][v] = ballot32(pred);  // [15:0]=row v, [31:16]=row v+8
      }
    }
    if (lane == 0) {
#pragma unroll
      for (int v = 0; v < 8; ++v) {
        int rA = r0 + v, rB = r0 + v + 8;
        unsigned wA = (bal[0][v] & 0xFFFFu) | ((bal[1][v] & 0xFFFFu) << 16);
        unsigned wB = (bal[0][v] >> 16) | ((bal[1][v] >> 16) << 16);
        if (rA < N) maskw[(size_t)rA * w32 + cg] = wA;
        if (rB < N) maskw[(size_t)rB * w32 + cg] = wB;
      }
    }
  }
}

// ---------------------------------------------------------------------------
// Kernel 5: sequential greedy suppression. Single wave32; removal mask lives
// in LDS; next mask row is double-buffered in registers to hide HBM latency.
// ---------------------------------------------------------------------------
__global__ __launch_bounds__(32) void nms_scan_kernel(const unsigned* __restrict__ maskw,
                                                      unsigned* __restrict__ keepw,
                                                      int N, int w32) {
  __shared__ unsigned rem[288];
  int lane = threadIdx.x;
  for (int w = lane; w < w32; w += 32) rem[w] = 0u;
  __syncthreads();
  unsigned cur[8], nxt[8];
#pragma unroll
  for (int k = 0; k < 8; ++k) {
    int w = lane + 32 * k;
    cur[k] = (w < w32) ? maskw[w] : 0u;
  }
  for (int i = 0; i < N; ++i) {
#pragma unroll
    for (int k = 0; k < 8; ++k) {
      int w = lane + 32 * k;
      nxt[k] = (w < w32 && (i + 1) < N) ? maskw[(size_t)(i + 1) * w32 + w] : 0u;
    }
    unsigned rm = rem[i >> 5];
    if (!((rm >> (i & 31)) & 1u)) {
#pragma unroll
      for (int k = 0; k < 8; ++k) {
        int w = lane + 32 * k;
        if (w < w32) rem[w] |= cur[k];
      }
    }
    __syncthreads();  // single-wave workgroup: lowers to S_NOP + counter waits
#pragma unroll
    for (int k = 0; k < 8; ++k) cur[k] = nxt[k];
  }
  for (int w = lane; w < w32; w += 32) {
    unsigned v = ~rem[w];
    int base = w * 32;
    int valid = N - base;  // >= 1 since w < w32
    if (valid < 32) v &= ((1u << valid) - 1u);
    keepw[w] = v;
  }
}

// ---------------------------------------------------------------------------
// Kernel 6: per-word prefix counts of kept boxes (184 words — one thread)
// ---------------------------------------------------------------------------
__global__ void prefix_kernel(const unsigned* __restrict__ keepw,
                              unsigned* __restrict__ prefix, int w32) {
  if (threadIdx.x == 0 && blockIdx.x == 0) {
    unsigned run = 0;
    for (int w = 0; w < w32; ++w) {
      prefix[w] = run;
      run += (unsigned)__popc(keepw[w]);
    }
  }
}

// ---------------------------------------------------------------------------
// Kernel 7a: zero output (invalid rows must be 0)
// ---------------------------------------------------------------------------
__global__ void zero_kernel(float* __restrict__ out, int n) {
  int i = blockIdx.x * blockDim.x + threadIdx.x;
  if (i < n) out[i] = 0.0f;
}

// ---------------------------------------------------------------------------
// Kernel 7b: scatter kept detections (rank via prefix + popcount)
// ---------------------------------------------------------------------------
__global__ void scatter_kernel(const unsigned* __restrict__ keepw,
                               const unsigned* __restrict__ prefix,
                               const unsigned* __restrict__ order,
                               const float* __restrict__ bbox,
                               const float* __restrict__ lm,
                               const float* __restrict__ scores,
                               const int* __restrict__ topk_p,
                               float* __restrict__ out, int N) {
  int i = blockIdx.x * blockDim.x + threadIdx.x;
  if (i >= N) return;
  unsigned wv = keepw[i >> 5];
  if (!((wv >> (i & 31)) & 1u)) return;
  int rank = (int)prefix[i >> 5] + __popc(wv & ((1u << (i & 31)) - 1u));
  int tk = *topk_p;
  if (rank >= tk) return;
  unsigned o = order[i];
  float* dr = out + (size_t)rank * 15;
  dr[0] = bbox[o * 4 + 0];
  dr[1] = bbox[o * 4 + 1];
  dr[2] = bbox[o * 4 + 2];
  dr[3] = bbox[o * 4 + 3];
#pragma unroll
  for (int t = 0; t < 10; ++t) dr[4 + t] = lm[o * 10 + t];
  dr[14] = scores[o];
}

// ---------------------------------------------------------------------------
extern "C" void kernel_launch(void* const* d_in, const int* in_sizes, int n_in,
                              void* d_out, int out_size, void* d_ws, size_t ws_size,
                              hipStream_t stream) {
  const float* loc = (const float*)d_in[0];
  const float* conf = (const float*)d_in[1];
  const float* iou = (const float*)d_in[2];
  const float* priors = (const float*)d_in[3];
  const int* topk_p = (const int*)d_in[4];
  float* out = (float*)d_out;

  int N = in_sizes[0] / 14;                 // 5875
  int apad = (N + 127) & ~127;              // 5888
  int w32 = (N + 31) / 32;                  // 184
  int sortn = 1;
  while (sortn < N) sortn <<= 1;            // 8192

  // Workspace layout (float units)
  float* wsf = (float*)d_ws;
  float* scores = wsf;                      // apad
  float* bbox = wsf + (size_t)apad;         // apad*4
  float* lm = wsf + (size_t)apad * 5;       // apad*10
  unsigned* order = (unsigned*)(wsf + (size_t)apad * 15);
  float* x1s = wsf + (size_t)apad * 16;
  float* y1s = wsf + (size_t)apad * 17;
  float* x2s = wsf + (size_t)apad * 18;
  float* y2s = wsf + (size_t)apad * 19;
  float* areas = wsf + (size_t)apad * 20;
  unsigned* keepw = (unsigned*)(wsf + (size_t)apad * 21);
  unsigned* prefix = (unsigned*)(wsf + (size_t)apad * 22);
  unsigned* maskw = (unsigned*)(wsf + (size_t)apad * 23);  // N*w32 words (~4.3MB)

  int nb = (N + 255) / 256;
  decode_kernel<<<nb, 256, 0, stream>>>(loc, conf, iou, priors, scores, bbox, lm, N);
  sort_kernel<<<1, 1024, (size_t)sortn * sizeof(unsigned long long), stream>>>(
      scores, order, N, sortn);
  gather_sorted_kernel<<<nb, 256, 0, stream>>>(order, bbox, x1s, y1s, x2s, y2s, areas, N);
  int rtiles = (N + 15) / 16;
  iou_mask_kernel<<<rtiles, 32, 0, stream>>>(x1s, y1s, x2s, y2s, areas, maskw, N, w32);
  nms_scan_kernel<<<1, 32, 0, stream>>>(maskw, keepw, N, w32);
  prefix_kernel<<<1, 32, 0, stream>>>(keepw, prefix, w32);
  zero_kernel<<<(out_size + 255) / 256, 256, 0, stream>>>(out, out_size);
  scatter_kernel<<<nb, 256, 0, stream>>>(keepw, prefix, order, bbox, lm, scores,
                                         topk_p, out, N);
}